// GIN_40767829574578
// MI455X (gfx1250) — compile-verified
//
#include <hip/hip_runtime.h>

#define GIN_N_NODES 100000
#define GIN_N_EDGES 1600000
#define GIN_K 128

typedef __attribute__((ext_vector_type(2))) float v2f;
typedef __attribute__((ext_vector_type(8))) float v8f;

// tmp = h  (identity term of GIN; edge contributions accumulate on top)
__global__ void gin_copy_kernel(const float* __restrict__ src,
                                float* __restrict__ dstp, int n4) {
  int i = blockIdx.x * blockDim.x + threadIdx.x;
  if (i < n4) ((float4*)dstp)[i] = ((const float4*)src)[i];
}

// One wave32 per edge: 32 lanes x float4 = 128 features, native f32 atomics.
__global__ void gin_scatter_kernel(const float* __restrict__ h,
                                   const int* __restrict__ src,
                                   const int* __restrict__ dst,
                                   float* __restrict__ tmp) {
  int edge = blockIdx.x * (blockDim.x >> 5) + (threadIdx.x >> 5);
  if (edge >= GIN_N_EDGES) return;
  int lane = threadIdx.x & 31;
  int s = src[edge];
  int d = dst[edge];
  float4 v = ((const float4*)(h + (size_t)s * GIN_K))[lane];
  float* p = tmp + (size_t)d * GIN_K + lane * 4;
  unsafeAtomicAdd(p + 0, v.x);
  unsafeAtomicAdd(p + 1, v.y);
  unsafeAtomicAdd(p + 2, v.z);
  unsafeAtomicAdd(p + 3, v.w);
}

// out[M x NOUT] = A[M x 128] * W[128 x NOUT] + bias, fp32 WMMA 16x16x4.
// Block = 128 threads = 4 waves; wave w owns 16-column tile blockIdx.y*4+w.
template <int NOUT>
__global__ void gin_gemm_wmma_kernel(const float* __restrict__ A,
                                     const float* __restrict__ W,
                                     const float* __restrict__ bias,
                                     float* __restrict__ out) {
  const int wave  = threadIdx.x >> 5;
  const int lane  = threadIdx.x & 31;
  const int mTile = blockIdx.x;
  const int nTile = blockIdx.y * 4 + wave;
  const int col   = nTile * 16 + (lane & 15);
  const int rowA  = mTile * 16 + (lane & 15);
  const int kOff  = (lane >> 4) * 2;   // lanes 16..31 carry K+2,K+3

  const float* Arow = A + (size_t)rowA * GIN_K + kOff;
  const float* Wcol = W + (size_t)kOff * NOUT + col;

  v8f c;
  const float bv = bias[col];
#pragma unroll
  for (int i = 0; i < 8; ++i) c[i] = bv;

#pragma unroll
  for (int kk = 0; kk < GIN_K; kk += 4) {
    v2f a = *(const v2f*)(Arow + kk);          // A[row][kk+kOff .. +1]
    v2f b;
    b[0] = Wcol[(size_t)kk * NOUT];            // W[kk+kOff][col]
    b[1] = Wcol[(size_t)(kk + 1) * NOUT];      // W[kk+kOff+1][col]
    c = __builtin_amdgcn_wmma_f32_16x16x4_f32(
        /*neg_a=*/false, a, /*neg_b=*/false, b,
        /*c_mod=*/(short)0, c, /*reuse_a=*/false, /*reuse_b=*/false);
  }

  // C/D layout: VGPR i -> row mTile*16 + i (+8 for lanes 16..31), col = lane&15.
  const int rowD = mTile * 16 + ((lane >> 4) << 3);
#pragma unroll
  for (int i = 0; i < 8; ++i) {
    out[(size_t)(rowD + i) * NOUT + col] = c[i];
  }
}

extern "C" void kernel_launch(void* const* d_in, const int* in_sizes, int n_in,
                              void* d_out, int out_size, void* d_ws,
                              size_t ws_size, hipStream_t stream) {
  const float* feat  = (const float*)d_in[0];
  const int*   src   = (const int*)d_in[1];
  const int*   dst   = (const int*)d_in[2];
  const float* W_in  = (const float*)d_in[3];
  const float* b_in  = (const float*)d_in[4];
  const float* W_hid = (const float*)d_in[5];
  const float* b_hid = (const float*)d_in[6];
  const float* W_out = (const float*)d_in[7];
  const float* b_out = (const float*)d_in[8];
  float* out = (float*)d_out;

  float* bufA = (float*)d_ws;                         // (1+A)h accumulator
  float* bufB = bufA + (size_t)GIN_N_NODES * GIN_K;   // post-linear activations

  const int  n4 = GIN_N_NODES * GIN_K / 4;
  const dim3 copyGrid((n4 + 255) / 256), copyBlk(256);
  const dim3 scatGrid(GIN_N_EDGES / 8), scatBlk(256);       // 8 waves/block
  const dim3 gemmGridH(GIN_N_NODES / 16, 2);                // N=128 -> 8 nTiles
  const dim3 gemmGridO(GIN_N_NODES / 16, 1);                // N=64  -> 4 nTiles
  const dim3 gemmBlk(128);

  // Layer 1: h1 = (feat + sum_E feat[src]) @ W_in + b_in
  gin_copy_kernel<<<copyGrid, copyBlk, 0, stream>>>(feat, bufA, n4);
  gin_scatter_kernel<<<scatGrid, scatBlk, 0, stream>>>(feat, src, dst, bufA);
  gin_gemm_wmma_kernel<128><<<gemmGridH, gemmBlk, 0, stream>>>(bufA, W_in, b_in, bufB);

  // Layer 2
  gin_copy_kernel<<<copyGrid, copyBlk, 0, stream>>>(bufB, bufA, n4);
  gin_scatter_kernel<<<scatGrid, scatBlk, 0, stream>>>(bufB, src, dst, bufA);
  gin_gemm_wmma_kernel<128><<<gemmGridH, gemmBlk, 0, stream>>>(bufA, W_hid, b_hid, bufB);

  // Layer 3 (output, 64 classes)
  gin_copy_kernel<<<copyGrid, copyBlk, 0, stream>>>(bufB, bufA, n4);
  gin_scatter_kernel<<<scatGrid, scatBlk, 0, stream>>>(bufB, src, dst, bufA);
  gin_gemm_wmma_kernel<64><<<gemmGridO, gemmBlk, 0, stream>>>(bufA, W_out, b_out, out);
}